// DenseDilatedKnnGraph_46007689675459
// MI455X (gfx1250) — compile-verified
//
#include <hip/hip_runtime.h>

typedef __attribute__((ext_vector_type(2))) float v2f;
typedef __attribute__((ext_vector_type(8))) float v8f;

#define BB 4
#define DD 64
#define NN 8192
#define KNN 9
#define ROWS_PER_WG 128
#define WAVES 8
#define COLS 32                 // columns staged per iteration (2 WMMA tiles wide)
#define NSTAGES (NN / COLS)
#define CB_STRIDE 66            // colbuf padded stride (floats)
#define SC_STRIDE 33            // scores padded stride (floats)

// ---------------- Kernel 1: L2-normalize points, produce (B,N,D) row-major + |p|^2 ----------------
__global__ __launch_bounds__(256) void normalize_kernel(const float* __restrict__ x,
                                                        float* __restrict__ pts,
                                                        float* __restrict__ sq) {
    const int t = blockIdx.x * 256 + threadIdx.x;   // [0, B*N)
    const int b = t >> 13;                          // / 8192
    const int n = t & (NN - 1);
    const float* xp = x + (size_t)b * DD * NN + n;  // x[b, d, n], stride N over d

    float xv[DD];
    float ss = 0.f;
#pragma unroll
    for (int d = 0; d < DD; ++d) { xv[d] = xp[(size_t)d * NN]; ss += xv[d] * xv[d]; }
    const float scale = 1.0f / fmaxf(sqrtf(ss), 1e-12f);

    float* pp = pts + (size_t)t * DD;
    float s2 = 0.f;
#pragma unroll
    for (int d = 0; d < DD; ++d) { const float v = xv[d] * scale; pp[d] = v; s2 += v * v; }
    sq[t] = s2;
}

// ---------------- Kernel 2: fused WMMA Gram tiles + full-wave streaming top-9 ----------------
__global__ __launch_bounds__(256) void knn_kernel(const float* __restrict__ pts,
                                                  const float* __restrict__ sq,
                                                  int* __restrict__ out) {
    __shared__ float colbuf[COLS * CB_STRIDE];            // 32 cols x 64 dims, padded
    __shared__ float sqbuf[COLS];
    __shared__ float scores[WAVES * 16 * SC_STRIDE];      // per-wave 16x32 key tile, padded
    __shared__ float mvals[WAVES * 16 * 18];              // final merge buffers
    __shared__ int   midxs[WAVES * 16 * 18];

    const int tid  = threadIdx.x;
    const int w    = tid >> 5;
    const int lane = tid & 31;
    const int b        = blockIdx.x >> 6;                 // 64 row-blocks per batch
    const int rowblock = blockIdx.x & 63;
    const int rowLocal = rowblock * ROWS_PER_WG + w * 16; // wave's first query row in batch

    // A fragments (16 rows x 64 K), loaded once, reused for all 256 stages.
    // ISA 32-bit A 16x4 layout: lane L -> M=L%16, K = 2*(L>=16) + {0,1}
    const int am = lane & 15;
    const int ak = (lane >> 4) << 1;
    const float* arow = pts + ((size_t)(b * NN + rowLocal + am)) * DD + ak;
    v2f afrag[16];
#pragma unroll
    for (int s = 0; s < 16; ++s)
        afrag[s] = *reinterpret_cast<const v2f*>(arow + 4 * s);

    // Per-lane private top-9: 2 lanes per row (lane, lane+16), each owns half the columns.
    float bestv[KNN];
    int   besti[KNN];
#pragma unroll
    for (int t = 0; t < KNN; ++t) { bestv[t] = -3.402823466e38f; besti[t] = 0; }

    const int scanRow  = lane & 15;
    const int scanHalf = lane >> 4;       // 0 -> cols 0..15 of stage, 1 -> cols 16..31

    for (int j = 0; j < NSTAGES; ++j) {
        const int col0 = j * COLS;
        __syncthreads();  // protect colbuf/sqbuf/scores reuse across waves
        {   // stage 32-column tile into LDS (8 floats per thread)
            const int idx = tid * 8;
            const int cc  = idx >> 6;
            const int dd  = idx & 63;
            const float* gsrc = pts + ((size_t)(b * NN + col0 + cc)) * DD + dd;
            const float4 v0 = *reinterpret_cast<const float4*>(gsrc);
            const float4 v1 = *reinterpret_cast<const float4*>(gsrc + 4);
            float* dst = &colbuf[cc * CB_STRIDE + dd];
            dst[0] = v0.x; dst[1] = v0.y; dst[2] = v0.z; dst[3] = v0.w;
            dst[4] = v1.x; dst[5] = v1.y; dst[6] = v1.z; dst[7] = v1.w;
            if (tid < COLS) sqbuf[tid] = sq[b * NN + col0 + tid];
            if (tid < 64 && j + 1 < NSTAGES)  // global_prefetch_b8 next stage (8 KB)
                __builtin_prefetch(pts + ((size_t)(b * NN + col0 + COLS)) * DD + tid * 32, 0, 1);
        }
        __syncthreads();

        // Two 16x16 Gram tiles: 32 x V_WMMA_F32_16X16X4_F32, two independent D-chains
        v8f acc0 = {};
        v8f acc1 = {};
#pragma unroll
        for (int s = 0; s < 16; ++s) {
            const v2f bf0 = *reinterpret_cast<const v2f*>(&colbuf[am * CB_STRIDE + 4 * s + ak]);
            const v2f bf1 = *reinterpret_cast<const v2f*>(&colbuf[(am + 16) * CB_STRIDE + 4 * s + ak]);
            acc0 = __builtin_amdgcn_wmma_f32_16x16x4_f32(false, afrag[s], false, bf0,
                                                         (short)0, acc0, false, false);
            acc1 = __builtin_amdgcn_wmma_f32_16x16x4_f32(false, afrag[s], false, bf1,
                                                         (short)0, acc1, false, false);
        }

        // fold |p_m|^2 and deposit keys: key = 2*dot - sq_m (max key == nearest)
        {
            const float sqc0  = sqbuf[am];
            const float sqc1  = sqbuf[am + 16];
            const int   rbase = (lane >> 4) << 3;          // rows 0..7 or 8..15
            float* srow = &scores[w * (16 * SC_STRIDE)];
#pragma unroll
            for (int v = 0; v < 8; ++v) {
                srow[(rbase + v) * SC_STRIDE + am]      = 2.0f * acc0[v] - sqc0;
                srow[(rbase + v) * SC_STRIDE + am + 16] = 2.0f * acc1[v] - sqc1;
            }
        }
        __syncthreads();

        // full-wave scan: 2 lanes per row, 16 candidates each (uniform control flow)
        {
            const float* sc = &scores[w * (16 * SC_STRIDE) + scanRow * SC_STRIDE + scanHalf * 16];
            const int mbase = col0 + scanHalf * 16;
#pragma unroll
            for (int c = 0; c < 16; ++c) {
                float v = sc[c];
                if (v > bestv[KNN - 1]) {                  // fast reject (common case)
                    int id = mbase + c;
#pragma unroll
                    for (int t = 0; t < KNN; ++t) {        // stable insert (strict >)
                        if (v > bestv[t]) {
                            const float tv = bestv[t]; const int ti = besti[t];
                            bestv[t] = v; besti[t] = id;
                            v = tv; id = ti;
                        }
                    }
                }
            }
        }
    }

    // ---- final merge of the two per-row sorted lists ----
    __syncthreads();
    {
        float* mv = &mvals[(w * 16 + scanRow) * 18 + scanHalf * 9];
        int*   mi = &midxs[(w * 16 + scanRow) * 18 + scanHalf * 9];
#pragma unroll
        for (int t = 0; t < KNN; ++t) { mv[t] = bestv[t]; mi[t] = besti[t]; }
    }
    __syncthreads();

    if (lane < 16) {
        const float* mv = &mvals[(w * 16 + lane) * 18];
        const int*   mi = &midxs[(w * 16 + lane) * 18];
        const int r = rowLocal + lane;
        const size_t o    = ((size_t)(b * NN + r)) * KNN;
        const size_t hout = (size_t)BB * NN * KNN;
        int i = 0, jj = 9;
#pragma unroll
        for (int t = 0; t < KNN; ++t) {    // 2-way merge, tie -> smaller index (top_k stability)
            const float av = mv[i], bv = mv[jj];
            const int   ai = mi[i], bi = mi[jj];
            const bool takeA = (av > bv) || ((av == bv) && (ai < bi));
            out[o + t]        = takeA ? ai : bi;  // nn_idx
            out[hout + o + t] = r;                // center_idx
            i  += takeA ? 1 : 0;
            jj += takeA ? 0 : 1;
        }
    }
}

extern "C" void kernel_launch(void* const* d_in, const int* in_sizes, int n_in,
                              void* d_out, int out_size, void* d_ws, size_t ws_size,
                              hipStream_t stream) {
    const float* x = (const float*)d_in[0];
    // k = 9, dilation = 1 are compile-time constants of this problem instance.
    float* pts = (float*)d_ws;                       // B*N*D floats = 8 MB
    float* sq  = pts + (size_t)BB * NN * DD;         // B*N floats
    int*   out = (int*)d_out;                        // (2, B, N, 9) int32

    normalize_kernel<<<(BB * NN) / 256, 256, 0, stream>>>(x, pts, sq);
    knn_kernel<<<BB * (NN / ROWS_PER_WG), 256, 0, stream>>>(pts, sq, out);
}